// VectorQuantizer_3968549781783
// MI455X (gfx1250) — compile-verified
//
#include <hip/hip_runtime.h>
#include <math.h>

typedef __attribute__((ext_vector_type(2))) float v2f;
typedef __attribute__((ext_vector_type(8))) float v8f;

#define N_E      1024
#define E_DIM    256
#define BETA     0.25f

// Output layout (floats), concatenated in reference return order:
//   z_q (16*256*64*64) | loss (1) | perplexity (1) | one_hot (65536*1024) | idx (65536) | d (65536*1024)
#define OFF_ZQ   0ull
#define OFF_LOSS 16777216ull
#define OFF_PERP 16777217ull
#define OFF_OH   16777218ull
#define OFF_IDX  83886082ull
#define OFF_D    83951618ull

// ws layout (floats): en2[1024] | hist[1024] | sse[1]

// ---------------------------------------------------------------------------
// Kernel 1: codebook norms + zero histogram/SSE (re-run every call)
// ---------------------------------------------------------------------------
__global__ void vq_init(const float* __restrict__ emb, float* __restrict__ ws) {
  int j = blockIdx.x * blockDim.x + threadIdx.x;
  if (j < N_E) {
    const float* e = emb + (size_t)j * E_DIM;
    float s = 0.f;
#pragma unroll 8
    for (int c = 0; c < E_DIM; ++c) { float v = e[c]; s += v * v; }
    ws[j] = s;            // en2
    ws[N_E + j] = 0.f;    // hist
  }
  if (j == 0) ws[2 * N_E] = 0.f;  // sse accumulator
}

// ---------------------------------------------------------------------------
// Kernel 2: fused distances (f32 WMMA) + argmin + d + one_hot + z_q + partials
// Block: 256 threads (8 waves), each block covers 128 rows of zf.
// ---------------------------------------------------------------------------
__launch_bounds__(256)
__global__ void vq_main(const float* __restrict__ z, const float* __restrict__ emb,
                        float* __restrict__ out, float* __restrict__ ws) {
  __shared__ float ldsB[16 * 260];   // emb column tile, pitch 260 -> conflict-free b64 reads
  __shared__ int   ldsIdx[128];      // best index per row of this block

  const int tid  = threadIdx.x;
  const int w    = tid >> 5;         // wave in block
  const int lane = tid & 31;
  const int r    = lane & 15;        // N index within tile / row-in-wave for A halves
  const int hi   = lane >> 4;        // 0: K pair {4t,4t+1}, 1: {4t+2,4t+3}

  const int m0  = blockIdx.x * 128;
  const int gmA = m0 + w * 16 + r;   // global zf row this lane loads halves of
  const int bb  = gmA >> 12;         // batch  (rows never cross batch: 128 | 4096)
  const int hw  = gmA & 4095;        // spatial position
  const float* zrow = z + (size_t)bb * 1048576u + (size_t)hw;

  // ---- A tile: 16 rows x 256 K, resident in registers (128 VGPRs/lane) ----
  float a[128];
#pragma unroll
  for (int t = 0; t < 64; ++t) {
    int k = 4 * t + 2 * hi;
    a[2 * t]     = zrow[(size_t)k * 4096u];
    a[2 * t + 1] = zrow[(size_t)(k + 1) * 4096u];
  }

  // ---- Row norms from the register tile ----
  float rn = 0.f;
#pragma unroll
  for (int i = 0; i < 128; ++i) rn += a[i] * a[i];
  rn += __shfl_xor(rn, 16);          // combine the two K-halves of each row
  float rn2v[8];
#pragma unroll
  for (int vg = 0; vg < 8; ++vg)     // map to C-layout row = vg + 8*hi
    rn2v[vg] = __shfl(rn, vg + (hi << 3));

  const float* en2 = ws;
  float* dout = out + OFF_D;

  float bestd[8];
  int   bidx[8];
#pragma unroll
  for (int vg = 0; vg < 8; ++vg) { bestd[vg] = 3.4e38f; bidx[vg] = 0; }

  // ---- Sweep the 64 codebook column tiles ----
  for (int nt = 0; nt < 64; ++nt) {
    const int n0 = nt * 16;
    {  // cooperative stage of emb[n0..n0+16) x 256 into LDS
      const int code = tid >> 4, seg = tid & 15;
      const float4* src = (const float4*)(emb + (size_t)(n0 + code) * 256 + seg * 16);
      float4* dst = (float4*)(&ldsB[code * 260 + seg * 16]);
      dst[0] = src[0]; dst[1] = src[1]; dst[2] = src[2]; dst[3] = src[3];
    }
    __syncthreads();

    // Two independent accumulator chains + batched B prefetch from LDS:
    // consecutive v_wmma ops are independent, and 4x ds_load_2addr_b64 are
    // outstanding before the first consumer waits.
    v8f acc0 = {}, acc1 = {};
    const float* brow = &ldsB[r * 260];
#pragma unroll
    for (int t8 = 0; t8 < 8; ++t8) {
      v2f bv[8];
#pragma unroll
      for (int u = 0; u < 8; ++u) {
        const int kk = 4 * (t8 * 8 + u) + 2 * hi;
        bv[u] = *(const v2f*)(brow + kk);              // ds_load_b64, conflict-free
      }
#pragma unroll
      for (int u = 0; u < 8; ++u) {
        const int t = t8 * 8 + u;
        v2f av; av[0] = a[2 * t]; av[1] = a[2 * t + 1];
        if (u & 1)
          acc1 = __builtin_amdgcn_wmma_f32_16x16x4_f32(
              false, av, false, bv[u], (short)0, acc1, false, false);
        else
          acc0 = __builtin_amdgcn_wmma_f32_16x16x4_f32(
              false, av, false, bv[u], (short)0, acc0, false, false);
      }
    }
    v8f acc = acc0 + acc1;

    const float en = en2[n0 + r];
#pragma unroll
    for (int vg = 0; vg < 8; ++vg) {
      const int rowm = m0 + w * 16 + vg + hi * 8;
      const float dv = rn2v[vg] + en - 2.0f * acc[vg];
      dout[(size_t)rowm * 1024u + (size_t)(n0 + r)] = dv;  // two 64B segments / store
      if (dv < bestd[vg]) { bestd[vg] = dv; bidx[vg] = n0 + r; }
    }
    __syncthreads();
  }

  // ---- Argmin butterfly across the 16 lanes sharing each row ----
#pragma unroll
  for (int vg = 0; vg < 8; ++vg) {
#pragma unroll
    for (int off = 8; off >= 1; off >>= 1) {
      float od = __shfl_xor(bestd[vg], off);
      int   oi = __shfl_xor(bidx[vg], off);
      if (od < bestd[vg] || (od == bestd[vg] && oi < bidx[vg])) {
        bestd[vg] = od; bidx[vg] = oi;
      }
    }
  }

  float* hist = ws + N_E;
  if (r == 0) {  // lanes 0 and 16: publish per-row results
#pragma unroll
    for (int vg = 0; vg < 8; ++vg) {
      const int rl = w * 16 + vg + hi * 8;
      ldsIdx[rl] = bidx[vg];
      out[OFF_IDX + (size_t)(m0 + rl)] = (float)bidx[vg];
      atomicAdd(&hist[bidx[vg]], 1.0f);
    }
  }
  __syncthreads();

  // ---- one_hot: zero this wave's 16 rows, then set the ones ----
  {
    float2* oh2 = (float2*)(out + OFF_OH + (size_t)(m0 + w * 16) * 1024u);
    float2 zz; zz.x = 0.f; zz.y = 0.f;
#pragma unroll 4
    for (int i = 0; i < 256; ++i) oh2[i * 32 + lane] = zz;  // in-order with the 1.0 below
  }
  const int myIdx = ldsIdx[w * 16 + r];
  if (lane < 16)
    out[OFF_OH + (size_t)(m0 + w * 16 + r) * 1024u + (size_t)myIdx] = 1.0f;

  // ---- z_q scatter (NCHW) + fused SSE for the loss ----
  float sse = 0.f;
  {
    const float* erow = emb + (size_t)myIdx * 256;
    float* zqout = out + OFF_ZQ;
    const size_t baseAddr = (size_t)bb * 1048576u + (size_t)hw;
#pragma unroll 4
    for (int c2 = 0; c2 < 128; ++c2) {
      const int c = 2 * c2 + hi;     // low lanes even channels, high lanes odd
      const float ev = erow[c];
      const size_t addr = baseAddr + (size_t)c * 4096u;
      const float zv = z[addr];
      zqout[addr] = ev;              // 16-float contiguous per half-wave
      const float df = ev - zv;
      sse += df * df;
    }
  }
#pragma unroll
  for (int off = 16; off >= 1; off >>= 1) sse += __shfl_xor(sse, off);
  if (lane == 0) atomicAdd(ws + 2 * N_E, sse);
}

// ---------------------------------------------------------------------------
// Kernel 3: finalize loss + perplexity
// ---------------------------------------------------------------------------
__global__ void vq_final(const float* __restrict__ ws, float* __restrict__ out) {
  __shared__ float red[256];
  const int t = threadIdx.x;
  float s = 0.f;
  for (int j = t; j < N_E; j += 256) {
    const float p = ws[N_E + j] * (1.0f / 65536.0f);
    s += p * logf(p + 1e-10f);
  }
  red[t] = s;
  __syncthreads();
  for (int o = 128; o >= 1; o >>= 1) {
    if (t < o) red[t] += red[t + o];
    __syncthreads();
  }
  if (t == 0) {
    out[OFF_PERP] = expf(-red[0]);
    out[OFF_LOSS] = (1.0f + BETA) * ws[2 * N_E] * (1.0f / 16777216.0f);
  }
}

// ---------------------------------------------------------------------------
extern "C" void kernel_launch(void* const* d_in, const int* in_sizes, int n_in,
                              void* d_out, int out_size, void* d_ws, size_t ws_size,
                              hipStream_t stream) {
  const float* z   = (const float*)d_in[0];   // (16,256,64,64)
  const float* emb = (const float*)d_in[1];   // (1024,256)
  float* out = (float*)d_out;
  float* ws  = (float*)d_ws;

  hipLaunchKernelGGL(vq_init,  dim3(4),   dim3(256), 0, stream, emb, ws);
  hipLaunchKernelGGL(vq_main,  dim3(512), dim3(256), 0, stream, z, emb, out, ws);
  hipLaunchKernelGGL(vq_final, dim3(1),   dim3(256), 0, stream, ws, out);
}